// RSSMCellDiscrete_35519379538349
// MI455X (gfx1250) — compile-verified
//
#include <hip/hip_runtime.h>
#include <hip/hip_bf16.h>

typedef float v2f __attribute__((ext_vector_type(2)));
typedef float v8f __attribute__((ext_vector_type(8)));
typedef int   v4i __attribute__((vector_size(16)));   // pointee type for async-LDS builtins

#define B_SZ   8192
#define EMBED  1536
#define ACT    6
#define DETER  2048
#define HID    2048
#define ZD     1024
#define G3     (3 * DETER)   // 6144
#define DISC   32
#define STOCH  32

#define BM 128
#define BN 64
#define BK 32
#define LDS_S 36   // 144B stride: 16B-aligned rows for b128, conflict-free frags

// gfx1250 async global->LDS path (guarded so compilation can't regress)
#if defined(__AMDGCN__) && __has_builtin(__builtin_amdgcn_global_load_async_to_lds_b128) && __has_builtin(__builtin_amdgcn_s_wait_asynccnt)
#define HAVE_ASYNC 1
#else
#define HAVE_ASYNC 0
#endif

// ---------------- Generic tiled GEMM: out = diag(mask) * (A @ W^T) + bias + addC
// A: [M,K] row-major, W: [N,K] row-major. M%128==0, N%64==0, K%32==0.
__global__ __launch_bounds__(256) void gemm_f32_wmma(
    const float* __restrict__ A, const float* __restrict__ W,
    const float* __restrict__ bias, const float* __restrict__ addC,
    const float* __restrict__ mask, float* __restrict__ out,
    int M, int N, int K)
{
    __shared__ float As[2][BM * LDS_S];
    __shared__ float Ws[2][BN * LDS_S];

    const int tid  = threadIdx.x;
    const int lane = tid & 31;
    const int wave = tid >> 5;     // 0..7
    const int wm   = wave >> 1;    // 0..3  (M direction)
    const int wn   = wave & 1;     // 0..1  (N direction)
    const int half = lane >> 4;    // 0/1
    const int lr   = lane & 15;

    const int bm = blockIdx.y * BM;
    const int bn = blockIdx.x * BN;
    const int nb = K / BK;

    // per-thread staging coordinates (fixed across K blocks)
    int arow[4], acol[4], wrow[2], wcol[2];
#pragma unroll
    for (int i = 0; i < 4; ++i) {
        int idx = tid + i * 256;
        arow[i] = idx >> 3;
        acol[i] = (idx & 7) << 2;
    }
#pragma unroll
    for (int i = 0; i < 2; ++i) {
        int idx = tid + i * 256;
        wrow[i] = idx >> 3;
        wcol[i] = (idx & 7) << 2;
    }

    v8f acc[2][2] = {};

    // compute one BK-block from LDS buffer p
    auto compute = [&](int p) {
        const float* Asb = &As[p][0];
        const float* Wsb = &Ws[p][0];
#pragma unroll
        for (int kk = 0; kk < BK; kk += 4) {
            const int kc = kk + 2 * half;   // lanes 0-15: K=kk,kk+1 ; 16-31: kk+2,kk+3
            v2f a[2], b[2];
#pragma unroll
            for (int t = 0; t < 2; ++t) {
                const float* pa = &Asb[(wm * 32 + t * 16 + lr) * LDS_S + kc];
                a[t][0] = pa[0]; a[t][1] = pa[1];
                const float* pb = &Wsb[(wn * 32 + t * 16 + lr) * LDS_S + kc];
                b[t][0] = pb[0]; b[t][1] = pb[1];
            }
#pragma unroll
            for (int i = 0; i < 2; ++i)
#pragma unroll
                for (int j = 0; j < 2; ++j)
                    acc[i][j] = __builtin_amdgcn_wmma_f32_16x16x4_f32(
                        false, a[i], false, b[j], (short)0, acc[i][j], false, false);
        }
    };

#if HAVE_ASYNC
    // ---- async global->LDS double-buffered pipeline (6 b128 copies / block) ----
    auto issue = [&](int ib, int p) {
        const int kb = ib * BK;
#pragma unroll
        for (int i = 0; i < 4; ++i) {
            const float* g = A + (size_t)(bm + arow[i]) * K + kb + acol[i];
            float* l = &As[p][arow[i] * LDS_S + acol[i]];
            __builtin_amdgcn_global_load_async_to_lds_b128((v4i*)g, (v4i*)l, 0, 0);
        }
#pragma unroll
        for (int i = 0; i < 2; ++i) {
            const float* g = W + (size_t)(bn + wrow[i]) * K + kb + wcol[i];
            float* l = &Ws[p][wrow[i] * LDS_S + wcol[i]];
            __builtin_amdgcn_global_load_async_to_lds_b128((v4i*)g, (v4i*)l, 0, 0);
        }
    };

    issue(0, 0);
    int p = 0;
    for (int ib = 0; ib < nb; ++ib) {
        if (ib + 1 < nb) {
            issue(ib + 1, p ^ 1);
            __builtin_amdgcn_s_wait_asynccnt(6);  // block ib's 6 copies retired (in-order)
        } else {
            __builtin_amdgcn_s_wait_asynccnt(0);
        }
        __syncthreads();          // whole WG's copies for buffer p visible
        compute(p);
        __syncthreads();          // release buffer p before it is refilled
        p ^= 1;
    }
#else
    // ---- fallback: register-prefetch double buffering ----
    float4 ra[4], rw[2];
    auto loadRegs = [&](int ib) {
        const int kb = ib * BK;
#pragma unroll
        for (int i = 0; i < 4; ++i)
            ra[i] = *(const float4*)(A + (size_t)(bm + arow[i]) * K + kb + acol[i]);
#pragma unroll
        for (int i = 0; i < 2; ++i)
            rw[i] = *(const float4*)(W + (size_t)(bn + wrow[i]) * K + kb + wcol[i]);
    };
    auto storeLDS = [&](int p) {
#pragma unroll
        for (int i = 0; i < 4; ++i) {
            float* d = &As[p][arow[i] * LDS_S + acol[i]];
            d[0] = ra[i].x; d[1] = ra[i].y; d[2] = ra[i].z; d[3] = ra[i].w;
        }
#pragma unroll
        for (int i = 0; i < 2; ++i) {
            float* d = &Ws[p][wrow[i] * LDS_S + wcol[i]];
            d[0] = rw[i].x; d[1] = rw[i].y; d[2] = rw[i].z; d[3] = rw[i].w;
        }
    };

    loadRegs(0);
    storeLDS(0);
    __syncthreads();
    int p = 0;
    for (int ib = 0; ib < nb; ++ib) {
        if (ib + 1 < nb) loadRegs(ib + 1);     // global loads overlap compute
        compute(p);
        if (ib + 1 < nb) {
            __syncthreads();
            storeLDS(p ^ 1);
            __syncthreads();
            p ^= 1;
        }
    }
#endif

    // ---- epilogue: row mask folded here (diag(mask)*AW^T == (mask.*A)W^T) ----
#pragma unroll
    for (int i = 0; i < 2; ++i) {
#pragma unroll
        for (int j = 0; j < 2; ++j) {
            int   gn = bn + wn * 32 + j * 16 + lr;
            float bv = bias ? bias[gn] : 0.0f;
#pragma unroll
            for (int e = 0; e < 8; ++e) {
                int    gm = bm + wm * 32 + i * 16 + half * 8 + e;
                size_t o  = (size_t)gm * N + gn;
                float  v  = acc[i][j][e];
                if (mask) v *= mask[gm];
                v += bv;
                if (addC) v += addC[o];
                out[o] = v;
            }
        }
    }
}

// ---------------- action @ a_mlp_w^T  (K = 6, trivial) ----------------
__global__ __launch_bounds__(256) void act_gemm_kernel(
    const float* __restrict__ action, const float* __restrict__ Wa,
    float* __restrict__ out)
{
    size_t idx = (size_t)blockIdx.x * 256 + threadIdx.x;  // over B*HID
    int m = (int)(idx >> 11);
    int n = (int)(idx & (HID - 1));
    const float* ar = action + (size_t)m * ACT;
    const float* wr = Wa + (size_t)n * ACT;
    float s = 0.f;
#pragma unroll
    for (int k = 0; k < ACT; ++k) s += ar[k] * wr[k];
    out[idx] = s;
}

// ---------------- LayerNorm + ELU over rows of width 2048 ----------------
__global__ __launch_bounds__(256) void ln_elu_kernel(
    const float* __restrict__ x, const float* __restrict__ g,
    const float* __restrict__ b, float* __restrict__ y)
{
    const int row = blockIdx.x;
    const float* xr = x + (size_t)row * HID;
    float* yr = y + (size_t)row * HID;
    __shared__ float red[256];

    float vals[8];
    float s = 0.f;
#pragma unroll
    for (int i = 0; i < 8; ++i) { vals[i] = xr[threadIdx.x + i * 256]; s += vals[i]; }
    red[threadIdx.x] = s; __syncthreads();
    for (int st = 128; st > 0; st >>= 1) {
        if (threadIdx.x < st) red[threadIdx.x] += red[threadIdx.x + st];
        __syncthreads();
    }
    float mu = red[0] * (1.0f / HID);
    __syncthreads();
    float vs = 0.f;
#pragma unroll
    for (int i = 0; i < 8; ++i) { float d = vals[i] - mu; vs += d * d; }
    red[threadIdx.x] = vs; __syncthreads();
    for (int st = 128; st > 0; st >>= 1) {
        if (threadIdx.x < st) red[threadIdx.x] += red[threadIdx.x + st];
        __syncthreads();
    }
    float inv = rsqrtf(red[0] * (1.0f / HID) + 1e-3f);
#pragma unroll
    for (int i = 0; i < 8; ++i) {
        int c = threadIdx.x + i * 256;
        float t = (vals[i] - mu) * inv * g[c] + b[c];
        yr[c] = t > 0.f ? t : expm1f(t);   // ELU
    }
}

// ---------------- GRU gate math (torch order r, z, n) ----------------
__global__ __launch_bounds__(256) void gru_gate_kernel(
    const float* __restrict__ gi, const float* __restrict__ gh,
    const float* __restrict__ in_h, const float* __restrict__ mask,
    float* __restrict__ h_out)
{
    size_t idx = (size_t)blockIdx.x * 256 + threadIdx.x;  // over B*DETER
    int m = (int)(idx >> 11);
    size_t base = (size_t)m * G3 + (idx & (DETER - 1));
    float ir = gi[base], iz = gi[base + DETER], in_ = gi[base + 2 * DETER];
    float hr = gh[base], hz = gh[base + DETER], hn  = gh[base + 2 * DETER];
    float h0 = in_h[idx] * mask[m];
    float r  = 1.0f / (1.0f + expf(-(ir + hr)));
    float zg = 1.0f / (1.0f + expf(-(iz + hz)));
    float nn = tanhf(in_ + r * hn);
    h_out[idx] = (1.0f - zg) * nn + zg * h0;
}

// ---------------- Gumbel-max straight-through sample ----------------
__global__ __launch_bounds__(256) void sample_kernel(
    const float* __restrict__ post, const float* __restrict__ noise,
    float* __restrict__ samp)
{
    size_t idx = (size_t)blockIdx.x * 256 + threadIdx.x;  // over B*STOCH
    size_t off = idx * DISC;
    const float* lg = post + off;
    const float* u  = noise + off;
    float l[32];
    float best = -3.4e38f; int bi = 0;
    float mx = -3.4e38f;
#pragma unroll
    for (int d = 0; d < DISC; ++d) {
        l[d] = lg[d];
        float uc = fminf(fmaxf(u[d], 1e-6f), 1.0f - 1e-6f);
        float gum = -logf(-logf(uc));
        float t = l[d] + gum;
        if (t > best) { best = t; bi = d; }
        mx = fmaxf(mx, l[d]);
    }
    float sum = 0.f;
#pragma unroll
    for (int d = 0; d < DISC; ++d) sum += expf(l[d] - mx);
    float rinv = 1.0f / sum;
    float* so = samp + off;
#pragma unroll
    for (int d = 0; d < DISC; ++d) {
        float p = expf(l[d] - mx) * rinv;
        float hard = (d == bi) ? 1.0f : 0.0f;
        so[d] = (hard - p) + p;   // straight-through value, as the reference computes it
    }
}

// ---------------- host orchestration ----------------
extern "C" void kernel_launch(void* const* d_in, const int* in_sizes, int n_in,
                              void* d_out, int out_size, void* d_ws, size_t ws_size,
                              hipStream_t stream)
{
    const float* embed    = (const float*)d_in[0];
    const float* action   = (const float*)d_in[1];
    const float* reset    = (const float*)d_in[2];
    const float* in_h     = (const float*)d_in[3];
    const float* in_z     = (const float*)d_in[4];
    const float* noise    = (const float*)d_in[5];
    const float* z_mlp_w  = (const float*)d_in[6];
    const float* z_mlp_b  = (const float*)d_in[7];
    const float* a_mlp_w  = (const float*)d_in[8];
    const float* in_ng    = (const float*)d_in[9];
    const float* in_nb    = (const float*)d_in[10];
    const float* gru_wih  = (const float*)d_in[11];
    const float* gru_whh  = (const float*)d_in[12];
    const float* gru_bih  = (const float*)d_in[13];
    const float* gru_bhh  = (const float*)d_in[14];
    const float* post_h_w = (const float*)d_in[15];
    const float* post_h_b = (const float*)d_in[16];
    const float* post_e_w = (const float*)d_in[17];
    const float* post_ng  = (const float*)d_in[18];
    const float* post_nb  = (const float*)d_in[19];
    const float* post_w   = (const float*)d_in[20];
    const float* post_b   = (const float*)d_in[21];

    float* out_post = (float*)d_out;                       // [B, 1024]
    float* out_h    = out_post + (size_t)B_SZ * ZD;        // [B, 2048]
    float* out_samp = out_post + (size_t)B_SZ * 3072;      // [B, 1024]

    float* ws     = (float*)d_ws;
    float* buf_x  = ws;                                    // [B, 2048]  x1 / x2
    float* buf_a  = buf_x + (size_t)B_SZ * HID;            // [B, 2048]  za / post_in
    float* buf_gi = buf_a + (size_t)B_SZ * HID;            // [B, 6144]  (ax early)
    float* buf_gh = buf_gi + (size_t)B_SZ * G3;            // [B, 6144]

    const dim3 blk(256);

    // 1) ax = action @ a_mlp_w^T  (stored in buf_gi, consumed before gi GEMM)
    act_gemm_kernel<<<(B_SZ * HID) / 256, blk, 0, stream>>>(action, a_mlp_w, buf_gi);

    // 2) x1 = (reset .* in_z) @ z_mlp_w^T + z_mlp_b + ax
    gemm_f32_wmma<<<dim3(HID / BN, B_SZ / BM), blk, 0, stream>>>(
        in_z, z_mlp_w, z_mlp_b, buf_gi, reset, buf_x, B_SZ, HID, ZD);

    // 3) za = ELU(LN(x1))
    ln_elu_kernel<<<B_SZ, blk, 0, stream>>>(buf_x, in_ng, in_nb, buf_a);

    // 4) gi = za @ gru_wih^T + gru_bih
    gemm_f32_wmma<<<dim3(G3 / BN, B_SZ / BM), blk, 0, stream>>>(
        buf_a, gru_wih, gru_bih, nullptr, nullptr, buf_gi, B_SZ, G3, HID);

    // 5) gh = (reset .* in_h) @ gru_whh^T + gru_bhh
    gemm_f32_wmma<<<dim3(G3 / BN, B_SZ / BM), blk, 0, stream>>>(
        in_h, gru_whh, gru_bhh, nullptr, reset, buf_gh, B_SZ, G3, DETER);

    // 6) GRU gates -> h  (written straight into d_out's h section)
    gru_gate_kernel<<<(B_SZ * DETER) / 256, blk, 0, stream>>>(
        buf_gi, buf_gh, in_h, reset, out_h);

    // 7) x2 = embed @ post_e_w^T
    gemm_f32_wmma<<<dim3(HID / BN, B_SZ / BM), blk, 0, stream>>>(
        embed, post_e_w, nullptr, nullptr, nullptr, buf_x, B_SZ, HID, EMBED);

    // 8) x2 += h @ post_h_w^T + post_h_b
    gemm_f32_wmma<<<dim3(HID / BN, B_SZ / BM), blk, 0, stream>>>(
        out_h, post_h_w, post_h_b, buf_x, nullptr, buf_x, B_SZ, HID, DETER);

    // 9) post_in = ELU(LN(x2))
    ln_elu_kernel<<<B_SZ, blk, 0, stream>>>(buf_x, post_ng, post_nb, buf_a);

    // 10) post = post_in @ post_w^T + post_b   -> d_out post section
    gemm_f32_wmma<<<dim3(ZD / BN, B_SZ / BM), blk, 0, stream>>>(
        buf_a, post_w, post_b, nullptr, nullptr, out_post, B_SZ, ZD, HID);

    // 11) straight-through Gumbel-max sample -> d_out sample section
    sample_kernel<<<(B_SZ * STOCH) / 256, blk, 0, stream>>>(out_post, noise, out_samp);
}